// HCF_21277267985141
// MI455X (gfx1250) — compile-verified
//
#include <hip/hip_runtime.h>

// ---------------------------------------------------------------------------
// HCF / LightGCN propagation for MI455X (gfx1250, wave32)
//   - on-device COO -> CSR-permutation build (hist + 1-block scan + scatter)
//   - atomic-free SpMM: 16 lanes/row, float4 (GLOBAL_LOAD_B128) gathers
//   - deterministic hash dropout
//   - layer mean via V_WMMA_F32_16X16X4_F32 (D = acc x (I/3))
// ---------------------------------------------------------------------------

typedef __attribute__((ext_vector_type(2))) float v2f;
typedef __attribute__((ext_vector_type(8))) float v8f;

#define TPB 256

__global__ void zero_i32_kernel(int* __restrict__ p, int n) {
    int i = blockIdx.x * blockDim.x + threadIdx.x;
    if (i < n) p[i] = 0;
}

__global__ void copy_f32_kernel(float* __restrict__ dst, const float* __restrict__ src, int n) {
    int i = blockIdx.x * blockDim.x + threadIdx.x;
    if (i < n) dst[i] = src[i];
}

__global__ void scale_f32_kernel(const float* __restrict__ src, float* __restrict__ dst,
                                 int n, float s) {
    int i = blockIdx.x * blockDim.x + threadIdx.x;
    if (i < n) dst[i] = src[i] * s;
}

// degree histogram of COO rows
__global__ void hist_kernel(const int* __restrict__ rows, int nnz, int* __restrict__ deg) {
    int e = blockIdx.x * blockDim.x + threadIdx.x;
    if (e < nnz) atomicAdd(&deg[rows[e]], 1);
}

// single-workgroup exclusive scan: deg[0..n) -> row_start[0..n], fill[0..n)
// (deg may alias fill: each chunk is read, synced, then overwritten)
__global__ void scan_kernel(const int* __restrict__ deg, int* __restrict__ row_start,
                            int* __restrict__ fill, int n) {
    __shared__ int sdata[1024];
    __shared__ int s_running;
    int tid = threadIdx.x;
    if (tid == 0) s_running = 0;
    __syncthreads();
    for (int base = 0; base < n; base += 1024) {
        int idx = base + tid;
        int v = (idx < n) ? deg[idx] : 0;
        sdata[tid] = v;
        __syncthreads();
        // Hillis-Steele inclusive scan
        for (int off = 1; off < 1024; off <<= 1) {
            int t2 = (tid >= off) ? sdata[tid - off] : 0;
            __syncthreads();
            sdata[tid] += t2;
            __syncthreads();
        }
        int incl = sdata[tid];
        int excl = incl - v;
        int run  = s_running;
        if (idx < n) {
            row_start[idx] = run + excl;
            fill[idx]      = run + excl;
        }
        __syncthreads();
        if (tid == 0) s_running = run + sdata[1023];
        __syncthreads();
    }
    if (threadIdx.x == 0) row_start[n] = s_running;
}

// scatter original nnz index into row-sorted permutation
__global__ void scatter_kernel(const int* __restrict__ rows, int nnz,
                               int* __restrict__ fill, int* __restrict__ perm) {
    int e = blockIdx.x * blockDim.x + threadIdx.x;
    if (e < nnz) {
        int pos = atomicAdd(&fill[rows[e]], 1);
        perm[pos] = e;
    }
}

// atomic-free CSR SpMM: y[r,:] = sum_j vals[e] * x[cols[e],:]
// 16 lanes per row, each lane owns 4 consecutive columns (float4 / B128).
__global__ void spmm_csr_kernel(const int* __restrict__ rs, const int* __restrict__ perm,
                                const int* __restrict__ cols, const float* __restrict__ vals,
                                const float* __restrict__ x, float* __restrict__ y, int nrows) {
    int gid = blockIdx.x * blockDim.x + threadIdx.x;
    int r = gid >> 4;
    int q = gid & 15;
    if (r >= nrows) return;
    const float4* x4 = (const float4*)x;
    int j0 = rs[r];
    int j1 = rs[r + 1];
    __builtin_prefetch((const void*)(perm + j0), 0, 1);  // global_prefetch_b8
    float4 acc = make_float4(0.f, 0.f, 0.f, 0.f);
    for (int j = j0; j < j1; ++j) {
        int e   = perm[j];
        float v = vals[e];
        int c   = cols[e];
        float4 xv = x4[(size_t)c * 16 + q];
        acc.x += v * xv.x;
        acc.y += v * xv.y;
        acc.z += v * xv.z;
        acc.w += v * xv.w;
    }
    ((float4*)y)[(size_t)r * 16 + q] = acc;
}

// deterministic hash-based dropout (x/(1-p) kept, 0 dropped) + layer accumulation
__global__ void dropout_acc_kernel(float* __restrict__ t, float* __restrict__ acc,
                                   int n, float keep, float inv_keep, unsigned seed) {
    int i = blockIdx.x * blockDim.x + threadIdx.x;
    if (i >= n) return;
    unsigned s = (unsigned)i ^ (seed * 0x9E3779B9u);
    s *= 0x85EBCA6Bu; s ^= s >> 13;
    s *= 0xC2B2AE35u; s ^= s >> 16;
    float u = (float)(s >> 8) * (1.0f / 16777216.0f);
    float v = (u < keep) ? t[i] * inv_keep : 0.0f;
    t[i] = v;
    acc[i] += v;
}

// out = acc * (1/3) expressed as 16x16 tile matmul against scaled identity:
//   D[16x16] = A[16x16] x (I/3), K split into 4 chunks of V_WMMA_F32_16X16X4_F32.
// One wave per 16x16 tile; EXEC stays all-ones (guard is wave-uniform).
__global__ void mean_wmma_kernel(const float* __restrict__ acc, float* __restrict__ out,
                                 int ntiles) {
    int wid  = (blockIdx.x * blockDim.x + threadIdx.x) >> 5;  // wave-uniform
    if (wid >= ntiles) return;
    int lane = threadIdx.x & 31;
    int tile_col = wid & 3;     // D = 64 -> 4 column tiles
    int tile_row = wid >> 2;
    int row0 = tile_row * 16;
    int col0 = tile_col * 16;
    int m    = lane & 15;       // A row index == B/D column index for this lane
    int half = lane >> 4;

    v8f c = {};
#pragma unroll
    for (int kc = 0; kc < 4; ++kc) {
        int k0 = kc * 4 + 2 * half;                       // K of VGPR0 slot for this lane
        const float* ap = acc + (size_t)(row0 + m) * 64 + col0 + k0;
        v2f a;
        a.x = ap[0];
        a.y = ap[1];
        v2f b;                                            // B[k][n] = (k==n)/3
        b.x = (k0     == m) ? (1.0f / 3.0f) : 0.0f;
        b.y = (k0 + 1 == m) ? (1.0f / 3.0f) : 0.0f;
        c = __builtin_amdgcn_wmma_f32_16x16x4_f32(
                /*neg_a=*/false, a, /*neg_b=*/false, b,
                /*c_mod=*/(short)0, c, /*reuse_a=*/false, /*reuse_b=*/false);
    }
#pragma unroll
    for (int v = 0; v < 8; ++v) {
        out[(size_t)(row0 + v + 8 * half) * 64 + col0 + m] = c[v];
    }
}

extern "C" void kernel_launch(void* const* d_in, const int* in_sizes, int n_in,
                              void* d_out, int out_size, void* d_ws, size_t ws_size,
                              hipStream_t stream) {
    const int D   = 64;
    const int N   = in_sizes[0] / D;
    const int NNZ = in_sizes[2];

    const float* user_emb = (const float*)d_in[0];
    const float* item_emb = (const float*)d_in[1];
    // matrices: 0=u1 1=u2 2=i1 3=i2 4=cat
    const int*   rows[5];
    const int*   cols[5];
    const float* vals[5];
    for (int m = 0; m < 5; ++m) {
        rows[m] = (const int*)d_in[2 + 3 * m];
        cols[m] = (const int*)d_in[3 + 3 * m];
        vals[m] = (const float*)d_in[4 + 3 * m];
    }

    // carve workspace (~111 MB for given sizes)
    char* p = (char*)d_ws;
    auto carve = [&](size_t bytes) -> char* {
        char* r = p;
        p += (bytes + 255) & ~(size_t)255;
        return r;
    };
    int* rs[5];
    int* perm[5];
    for (int m = 0; m < 5; ++m) rs[m]   = (int*)carve((size_t)(N + 1) * 4);
    for (int m = 0; m < 5; ++m) perm[m] = (int*)carve((size_t)NNZ * 4);
    int*   fill = (int*)carve((size_t)N * 4);
    float* acc  = (float*)carve((size_t)N * D * 4);
    float* ta   = (float*)carve((size_t)N * D * 4);
    float* tb   = (float*)carve((size_t)N * D * 4);
    (void)ws_size; (void)n_in; (void)out_size;

    const int nd   = N * D;
    const int gN   = (N + TPB - 1) / TPB;
    const int gE   = (NNZ + TPB - 1) / TPB;
    const int gND  = (nd + TPB - 1) / TPB;
    const int gROW = (N * 16 + TPB - 1) / TPB;

    // ---- build CSR permutations for all 5 matrices ----
    for (int m = 0; m < 5; ++m) {
        zero_i32_kernel<<<gN, TPB, 0, stream>>>(fill, N);
        hist_kernel<<<gE, TPB, 0, stream>>>(rows[m], NNZ, fill);
        scan_kernel<<<1, 1024, 0, stream>>>(fill, rs[m], fill, N);
        scatter_kernel<<<gE, TPB, 0, stream>>>(rows[m], NNZ, fill, perm[m]);
    }

    auto spmm = [&](int m, const float* x, float* y) {
        spmm_csr_kernel<<<gROW, TPB, 0, stream>>>(rs[m], perm[m], cols[m], vals[m], x, y, N);
    };

    float* outp = (float*)d_out;
    const float keep = 0.9f;               // P = (0.1, 0.1)
    const float inv_keep = 1.0f / keep;

    for (int path = 0; path < 2; ++path) {
        const float* E = (path == 0) ? user_emb : item_emb;
        const int a2 = (path == 0) ? 1 : 3;  // u2 / i2
        const int a1 = (path == 0) ? 0 : 2;  // u1 / i1
        const int tag = path;

        copy_f32_kernel<<<gND, TPB, 0, stream>>>(acc, E, nd);  // acc = h0

        // layer 0: t = cat @ (a1 @ (a2 @ h0)); dropout; acc += h1
        spmm(a2, E, ta);
        spmm(a1, ta, tb);
        spmm(4, tb, ta);
        dropout_acc_kernel<<<gND, TPB, 0, stream>>>(ta, acc, nd, keep, inv_keep,
                                                    (unsigned)(tag * 2 + 0 + 1));

        // layer 1 (input h1 = ta)
        spmm(a2, ta, tb);
        spmm(a1, tb, ta);
        spmm(4, ta, tb);
        dropout_acc_kernel<<<gND, TPB, 0, stream>>>(tb, acc, nd, keep, inv_keep,
                                                    (unsigned)(tag * 2 + 1 + 1));

        // mean over {h0,h1,h2}: out = acc / 3 via WMMA identity matmul
        float* OUT = outp + (size_t)path * nd;
        if ((N & 15) == 0) {
            int ntiles  = (N / 16) * 4;
            int threads = ntiles * 32;
            int blocks  = (threads + TPB - 1) / TPB;
            mean_wmma_kernel<<<blocks, TPB, 0, stream>>>(acc, OUT, ntiles);
        } else {
            scale_f32_kernel<<<gND, TPB, 0, stream>>>(acc, OUT, nd, 1.0f / 3.0f);
        }
    }
}